// HeteroConvLayerCausalCus_73023033966984
// MI455X (gfx1250) — compile-verified
//
#include <hip/hip_runtime.h>

typedef __attribute__((ext_vector_type(2))) float v2f;
typedef __attribute__((ext_vector_type(8))) float v8f;
typedef __attribute__((ext_vector_type(4))) int   v4i_t;
typedef __attribute__((address_space(1))) v4i_t*  gv4i_p;   // global int4*
typedef __attribute__((address_space(3))) v4i_t*  lv4i_p;   // LDS int4*

#define DIN 300
#define DOUT 128
#define W_CHUNKS ((DIN * DOUT) / 4)   // float4 chunks of W

// Rotation swizzle for W staged in LDS: row k holds its 128 cols rotated by
// (k>>1)*16. WMMA B-fragment reads (lanes 0-15 -> row K, lanes 16-31 -> row
// K+2) then hit disjoint 16-bank groups => conflict-free, zero padding.
__device__ __forceinline__ int ldsIdx(int k, int col) {
    return k * DOUT + ((col + ((k >> 1) << 4)) & (DOUT - 1));
}

// ---------------------------------------------------------------------------
// GEMM + bias:  out[M,128] = A[M,300] @ W[300,128] + bias[128]
// One wave per 16-row tile, 8 v8f accumulators = all 128 columns.
// W is staged once per block into LDS (153.6 KB; 2 blocks/WGP in 320 KB LDS)
// via GLOBAL_LOAD_ASYNC_TO_LDS_B128 (ASYNCcnt path, LDS written directly,
// swizzle applied through the per-lane LDS destination address), then the
// K-loop feeds conflict-free ds_loads into V_WMMA_F32_16X16X4_F32.
// ---------------------------------------------------------------------------
__global__ __launch_bounds__(256) void gemm_bias_wmma(
    const float* __restrict__ A, const float* __restrict__ W,
    const float* __restrict__ bias, float* __restrict__ out, int M)
{
    extern __shared__ float lds[];   // DIN * DOUT floats = 153600 bytes

#if __has_builtin(__builtin_amdgcn_global_load_async_to_lds_b128)
    // Async staging: memory -> LDS directly (no VGPR round-trip).
    for (int i = threadIdx.x; i < W_CHUNKS; i += 256) {
        const int row = i >> 5;          // float4-index: 32 per row
        const int c0  = (i & 31) << 2;   // start column
        __builtin_amdgcn_global_load_async_to_lds_b128(
            (gv4i_p)(W + (size_t)i * 4),
            (lv4i_p)&lds[ldsIdx(row, c0)],
            0, 0);
    }
    asm volatile("s_wait_asynccnt 0x0" ::: "memory");
#else
    // Fallback: cooperative load through VGPRs.
    for (int i = threadIdx.x; i < W_CHUNKS; i += 256) {
        const int row = i >> 5;
        const int c0  = (i & 31) << 2;
        const float4 v = ((const float4*)W)[i];
        *(float4*)(&lds[ldsIdx(row, c0)]) = v;
    }
#endif
    __syncthreads();

    const int lane = threadIdx.x & 31;
    const int wave = threadIdx.x >> 5;
    const int tile = blockIdx.x * 8 + wave;
    const int row0 = tile * 16;
    if (row0 >= M) return;               // wave-uniform exit (EXEC full at WMMA)

    const int mrow   = row0 + (lane & 15);
    const int mclamp = (mrow < M) ? mrow : (M - 1);  // clamp: bad rows never stored
    const int khalf  = (lane >> 4) * 2;  // lanes 0-15 -> K{0,1}; 16-31 -> K{2,3}
    const int ncol   = lane & 15;

    v8f c[8];
#pragma unroll
    for (int n = 0; n < 8; ++n) c[n] = (v8f){0.f,0.f,0.f,0.f,0.f,0.f,0.f,0.f};

    const float* arow = A + (size_t)mclamp * DIN;

    for (int kk = 0; kk < DIN; kk += 4) {
        const int k0 = kk + khalf;
        const v2f a = *(const v2f*)(arow + k0);     // one b64 load, branch-free
#pragma unroll
        for (int n = 0; n < 8; ++n) {
            v2f b;
            b.x = lds[ldsIdx(k0,     n * 16 + ncol)];
            b.y = lds[ldsIdx(k0 + 1, n * 16 + ncol)];
            c[n] = __builtin_amdgcn_wmma_f32_16x16x4_f32(
                false, a, false, b, (short)0, c[n], false, false);
        }
    }

    // C/D layout: VGPR r -> row (r + 8*(lane>=16)), lane&15 -> column
    const int rbase = row0 + ((lane >> 4) << 3);
#pragma unroll
    for (int n = 0; n < 8; ++n) {
        const int col = n * 16 + ncol;
        const float bb = bias[col];
#pragma unroll
        for (int r = 0; r < 8; ++r) {
            const int rr = rbase + r;
            if (rr < M) out[(size_t)rr * DOUT + col] = c[n][r] + bb;
        }
    }
}

// ---------------------------------------------------------------------------
// Edge scatter: one wave per edge; each lane carries a float4 (128 cols / 32
// lanes). Wh rows + accumulators are L2-resident (78 MB << 192 MB L2), so the
// global_atomic_add_f32 traffic stays at L2 throughput, never HBM RMW.
// ---------------------------------------------------------------------------
__global__ __launch_bounds__(256) void scatter_edges(
    const float* __restrict__ Wh, const int* __restrict__ src,
    const int* __restrict__ dst, const float* __restrict__ w,
    float* __restrict__ acc, float* __restrict__ cnt, int nE)
{
    const long long t = (long long)blockIdx.x * blockDim.x + threadIdx.x;
    const int e    = (int)(t >> 5);      // wave-uniform
    const int lane = (int)(t & 31);
    if (e >= nE) return;
    const int   s  = src[e];
    const int   d  = dst[e];
    const float we = w[e];
    const float4 v = ((const float4*)(Wh + (size_t)s * DOUT))[lane];
    float* o = acc + (size_t)d * DOUT + lane * 4;
    atomicAdd(o + 0, v.x * we);
    atomicAdd(o + 1, v.y * we);
    atomicAdd(o + 2, v.z * we);
    atomicAdd(o + 3, v.w * we);
    if (lane == 0) atomicAdd(cnt + d, 1.0f);
}

// out = acc / max(cnt,1)
__global__ __launch_bounds__(256) void finalize1(
    const float* __restrict__ acc, const float* __restrict__ cnt,
    float* __restrict__ out, int n)
{
    const int t = blockIdx.x * blockDim.x + threadIdx.x;
    if (t >= n * DOUT) return;
    out[t] = acc[t] / fmaxf(cnt[t >> 7], 1.0f);
}

// out = accA/max(cntA,1) + accB/max(cntB,1)
__global__ __launch_bounds__(256) void finalize2(
    const float* __restrict__ accA, const float* __restrict__ cntA,
    const float* __restrict__ accB, const float* __restrict__ cntB,
    float* __restrict__ out, int n)
{
    const int t = blockIdx.x * blockDim.x + threadIdx.x;
    if (t >= n * DOUT) return;
    const int i = t >> 7;
    out[t] = accA[t] / fmaxf(cntA[i], 1.0f) + accB[t] / fmaxf(cntB[i], 1.0f);
}

__global__ __launch_bounds__(256) void zero_f32(float* __restrict__ p, long long n)
{
    long long t = (long long)blockIdx.x * blockDim.x + threadIdx.x;
    const long long stride = (long long)gridDim.x * blockDim.x;
    for (; t < n; t += stride) p[t] = 0.0f;
}

// ---------------------------------------------------------------------------
extern "C" void kernel_launch(void* const* d_in, const int* in_sizes, int n_in,
                              void* d_out, int out_size, void* d_ws, size_t ws_size,
                              hipStream_t stream)
{
    (void)in_sizes; (void)n_in; (void)out_size; (void)ws_size;

    const int N_WORD = 30000, N_TOPIC = 1000, N_DOC = 15000;
    const int E_ww = 800000, E_wt = 400000, E_wd = 600000, E_td = 300000, E_tt = 150000;

    // Inputs in setup_inputs() dict order.
    const float* feat_word  = (const float*)d_in[0];
    const float* feat_topic = (const float*)d_in[1];
    const int*   ww_src = (const int*)d_in[2];
    const int*   ww_dst = (const int*)d_in[3];
    const float* ww_w   = (const float*)d_in[4];
    const float* W_ww   = (const float*)d_in[5];
    const float* b_ww   = (const float*)d_in[6];
    const int*   wt_src = (const int*)d_in[7];
    const int*   wt_dst = (const int*)d_in[8];
    const float* wt_w   = (const float*)d_in[9];
    const float* W_wt   = (const float*)d_in[10];
    const float* b_wt   = (const float*)d_in[11];
    const int*   wd_src = (const int*)d_in[12];
    const int*   wd_dst = (const int*)d_in[13];
    const float* wd_w   = (const float*)d_in[14];
    const float* W_wd   = (const float*)d_in[15];
    const float* b_wd   = (const float*)d_in[16];
    const int*   td_src = (const int*)d_in[17];
    const int*   td_dst = (const int*)d_in[18];
    const float* td_w   = (const float*)d_in[19];
    const float* W_td   = (const float*)d_in[20];
    const float* b_td   = (const float*)d_in[21];
    const int*   tt_src = (const int*)d_in[22];
    const int*   tt_dst = (const int*)d_in[23];
    const float* tt_w   = (const float*)d_in[24];
    const float* W_tt   = (const float*)d_in[25];
    const float* b_tt   = (const float*)d_in[26];

    // Workspace layout (floats)
    float* ws = (float*)d_ws;
    size_t off = 0;
    float* Wh_ww = ws + off; off += (size_t)N_WORD  * DOUT;
    float* Wh_wt = ws + off; off += (size_t)N_WORD  * DOUT;
    float* Wh_wd = ws + off; off += (size_t)N_WORD  * DOUT;
    float* Wh_td = ws + off; off += (size_t)N_TOPIC * DOUT;
    float* Wh_tt = ws + off; off += (size_t)N_TOPIC * DOUT;
    float* zero_base = ws + off;          // everything below re-zeroed per call
    float* acc_ww = ws + off; off += (size_t)N_WORD  * DOUT;
    float* acc_wt = ws + off; off += (size_t)N_TOPIC * DOUT;
    float* acc_tt = ws + off; off += (size_t)N_TOPIC * DOUT;
    float* acc_wd = ws + off; off += (size_t)N_DOC   * DOUT;
    float* acc_td = ws + off; off += (size_t)N_DOC   * DOUT;
    float* cnt_ww = ws + off; off += N_WORD;
    float* cnt_wt = ws + off; off += N_TOPIC;
    float* cnt_tt = ws + off; off += N_TOPIC;
    float* cnt_wd = ws + off; off += N_DOC;
    float* cnt_td = ws + off; off += N_DOC;
    const long long zero_count = (long long)((ws + off) - zero_base);

    // Output layout: h_word | h_topic | h_doc
    float* h_word  = (float*)d_out;
    float* h_topic = h_word  + (size_t)N_WORD  * DOUT;
    float* h_doc   = h_topic + (size_t)N_TOPIC * DOUT;

    // 1) zero accumulators + counts
    zero_f32<<<2048, 256, 0, stream>>>(zero_base, zero_count);

    // 2) WMMA GEMMs: Wh = feat @ W + b  (153.6 KB dynamic LDS for W staging)
    const size_t ldsBytes = (size_t)DIN * DOUT * sizeof(float);
    auto gemm_blocks = [](int M) { return ((M + 15) / 16 + 7) / 8; };
    gemm_bias_wmma<<<gemm_blocks(N_WORD),  256, ldsBytes, stream>>>(feat_word,  W_ww, b_ww, Wh_ww, N_WORD);
    gemm_bias_wmma<<<gemm_blocks(N_WORD),  256, ldsBytes, stream>>>(feat_word,  W_wt, b_wt, Wh_wt, N_WORD);
    gemm_bias_wmma<<<gemm_blocks(N_WORD),  256, ldsBytes, stream>>>(feat_word,  W_wd, b_wd, Wh_wd, N_WORD);
    gemm_bias_wmma<<<gemm_blocks(N_TOPIC), 256, ldsBytes, stream>>>(feat_topic, W_td, b_td, Wh_td, N_TOPIC);
    gemm_bias_wmma<<<gemm_blocks(N_TOPIC), 256, ldsBytes, stream>>>(feat_topic, W_tt, b_tt, Wh_tt, N_TOPIC);

    // 3) Edge scatters (L2-resident f32 atomics), one wave per edge
    auto sc_blocks = [](int nE) { return (int)(((long long)nE * 32 + 255) / 256); };
    scatter_edges<<<sc_blocks(E_ww), 256, 0, stream>>>(Wh_ww, ww_src, ww_dst, ww_w, acc_ww, cnt_ww, E_ww);
    scatter_edges<<<sc_blocks(E_wt), 256, 0, stream>>>(Wh_wt, wt_src, wt_dst, wt_w, acc_wt, cnt_wt, E_wt);
    scatter_edges<<<sc_blocks(E_wd), 256, 0, stream>>>(Wh_wd, wd_src, wd_dst, wd_w, acc_wd, cnt_wd, E_wd);
    scatter_edges<<<sc_blocks(E_td), 256, 0, stream>>>(Wh_td, td_src, td_dst, td_w, acc_td, cnt_td, E_td);
    scatter_edges<<<sc_blocks(E_tt), 256, 0, stream>>>(Wh_tt, tt_src, tt_dst, tt_w, acc_tt, cnt_tt, E_tt);

    // 4) Finalize: per-etype mean, summed per destination node type
    finalize1<<<(N_WORD * DOUT + 255) / 256, 256, 0, stream>>>(acc_ww, cnt_ww, h_word, N_WORD);
    finalize2<<<(N_TOPIC * DOUT + 255) / 256, 256, 0, stream>>>(acc_wt, cnt_wt, acc_tt, cnt_tt, h_topic, N_TOPIC);
    finalize2<<<(N_DOC * DOUT + 255) / 256, 256, 0, stream>>>(acc_wd, cnt_wd, acc_td, cnt_td, h_doc, N_DOC);
}